// IHT_56573309224252
// MI455X (gfx1250) — compile-verified
//
#include <hip/hip_runtime.h>

// ---------------------------------------------------------------------------
// IHT on MI455X (gfx1250): fp32 WMMA (V_WMMA_F32_16X16X4_F32) GEMMs.
// A (128MB) is L2-resident (192MB L2), problem is bandwidth bound (~17 FLOP/B),
// so we keep full fp32 precision for bit-faithful top-k selection.
// ---------------------------------------------------------------------------

typedef float v2f __attribute__((ext_vector_type(2)));
typedef float v8f __attribute__((ext_vector_type(8)));

#define BATCH     32
#define M_DIM     2048
#define N_DIM     16384
#define SPARSITY  64
#define STEP      0.5f
#define MAX_ITERS 20
#define TOL       1e-6f
#define KSPLIT    8

// ---------------- init: x = 0, done = 0 -----------------------------------
__global__ void iht_init(float* __restrict__ x, int* __restrict__ done) {
    int i = blockIdx.x * blockDim.x + threadIdx.x;
    if (i < BATCH * N_DIM) x[i] = 0.0f;
    if (i < BATCH) done[i] = 0;
}

// ---------------- K1: partial (x @ A^T) -----------------------------------
// grid = 128 m-tiles * KSPLIT k-chunks, one wave32 per block.
// WMMA: D(16b x 16m) = Xtile(16b x 4k) * A^T tile(4k x 16m).
// Both operands: contiguous float2 per lane (K is the fast axis of x and A).
__global__ __launch_bounds__(32) void iht_xat(const float* __restrict__ x,
                                              const float* __restrict__ A,
                                              float* __restrict__ Rpart) {
    const int mtile = blockIdx.x & 127;
    const int kc    = blockIdx.x >> 7;
    const int lane  = threadIdx.x;
    const int hi    = lane >> 4;      // 0: K=0,1  1: K=2,3
    const int lo    = lane & 15;
    const int m0    = mtile * 16;

    v8f acc0 = {};   // batch rows 0..15
    v8f acc1 = {};   // batch rows 16..31

    const float* xrow0 = x + (size_t)lo * N_DIM;
    const float* xrow1 = x + (size_t)(lo + 16) * N_DIM;
    const float* arow  = A + (size_t)(m0 + lo) * N_DIM;

    const int kbeg = kc * (N_DIM / KSPLIT);
    const int kend = kbeg + (N_DIM / KSPLIT);
    for (int k0 = kbeg; k0 < kend; k0 += 4) {
        const int off = k0 + 2 * hi;
        v2f a0 = *(const v2f*)(xrow0 + off);   // A-op tile, batch 0..15
        v2f a1 = *(const v2f*)(xrow1 + off);   // A-op tile, batch 16..31
        v2f b  = *(const v2f*)(arow  + off);   // B-op tile (A^T): A[m0+lane, k0+2hi..+1]
        acc0 = __builtin_amdgcn_wmma_f32_16x16x4_f32(false, a0, false, b,
                                                     (short)0, acc0, false, false);
        acc1 = __builtin_amdgcn_wmma_f32_16x16x4_f32(false, a1, false, b,
                                                     (short)0, acc1, false, false);
    }

    // C/D layout: VGPR v, lanes 0-15 -> M=v ; lanes 16-31 -> M=v+8 ; N=lane&15
    float* out = Rpart + (size_t)kc * BATCH * M_DIM;
    const int mcol = m0 + lo;
    #pragma unroll
    for (int v = 0; v < 8; v++) {
        const int brow = v + 8 * hi;
        out[(size_t)brow * M_DIM + mcol]        = acc0[v];
        out[(size_t)(brow + 16) * M_DIM + mcol] = acc1[v];
    }
}

// ---------------- K1b: R = y - sum_k Rpart --------------------------------
__global__ void iht_resid(const float* __restrict__ y,
                          const float* __restrict__ Rpart,
                          float* __restrict__ R) {
    int i = blockIdx.x * blockDim.x + threadIdx.x;
    if (i >= BATCH * M_DIM) return;
    float s = 0.0f;
    #pragma unroll
    for (int c = 0; c < KSPLIT; c++) s += Rpart[(size_t)c * BATCH * M_DIM + i];
    R[i] = y[i] - s;
}

// ---------------- K2: xh = x + STEP * (R @ A) ------------------------------
// grid = 1024 n-tiles, one wave32 per block, full K=2048 per wave.
// B-op loads of A are lane-contiguous (coalesced 64B segments).
__global__ __launch_bounds__(32) void iht_grad(const float* __restrict__ R,
                                               const float* __restrict__ A,
                                               const float* __restrict__ x,
                                               float* __restrict__ xh) {
    const int n0   = blockIdx.x * 16;
    const int lane = threadIdx.x;
    const int hi   = lane >> 4;
    const int lo   = lane & 15;

    v8f acc0 = {};
    v8f acc1 = {};

    const float* rrow0 = R + (size_t)lo * M_DIM;
    const float* rrow1 = R + (size_t)(lo + 16) * M_DIM;

    for (int m0 = 0; m0 < M_DIM; m0 += 4) {
        const int off = m0 + 2 * hi;
        v2f a0 = *(const v2f*)(rrow0 + off);
        v2f a1 = *(const v2f*)(rrow1 + off);
        v2f b;
        b.x = A[(size_t)off * N_DIM + n0 + lo];        // K = 2hi   (VGPR0)
        b.y = A[(size_t)(off + 1) * N_DIM + n0 + lo];  // K = 2hi+1 (VGPR1)
        acc0 = __builtin_amdgcn_wmma_f32_16x16x4_f32(false, a0, false, b,
                                                     (short)0, acc0, false, false);
        acc1 = __builtin_amdgcn_wmma_f32_16x16x4_f32(false, a1, false, b,
                                                     (short)0, acc1, false, false);
    }

    const int n = n0 + lo;
    #pragma unroll
    for (int v = 0; v < 8; v++) {
        const int b0 = v + 8 * hi;
        xh[(size_t)b0 * N_DIM + n]        = x[(size_t)b0 * N_DIM + n]        + STEP * acc0[v];
        xh[(size_t)(b0 + 16) * N_DIM + n] = x[(size_t)(b0 + 16) * N_DIM + n] + STEP * acc1[v];
    }
}

// ---------------- K3: top-64 hard threshold + done mask --------------------
// One block per batch row. Exact top-k via binary search on |v| bit pattern
// (monotone for non-negative floats); ties resolved to keep exactly k.
__global__ __launch_bounds__(256) void iht_topk(const float* __restrict__ R,
                                                const float* __restrict__ xh,
                                                float* __restrict__ x,
                                                int* __restrict__ done) {
    __shared__ int   sred[256];
    __shared__ float fred[256];
    __shared__ unsigned sctr;

    const int b = blockIdx.x;
    const int t = threadIdx.x;

    // residual norm^2 (uses residual of THIS iteration, pre-update)
    const float* rrow = R + (size_t)b * M_DIM;
    float s = 0.0f;
    for (int i = t; i < M_DIM; i += 256) { float v = rrow[i]; s += v * v; }
    fred[t] = s; __syncthreads();
    for (int w = 128; w > 0; w >>= 1) { if (t < w) fred[t] += fred[t + w]; __syncthreads(); }
    const float norm2 = fred[0];
    const int done_old = done[b];
    __syncthreads();

    if (!done_old) {
        const float* row = xh + (size_t)b * N_DIM;

        // t* = max t such that count(|v| >= t) >= k   (on uint key space)
        unsigned klo = 0u, khi = 0x7f800000u;
        while (klo < khi) {
            const unsigned mid = klo + ((khi - klo + 1u) >> 1);
            int cnt = 0;
            for (int i = t; i < N_DIM; i += 256) {
                unsigned key = __float_as_uint(row[i]) & 0x7fffffffu;
                cnt += (key >= mid);
            }
            sred[t] = cnt; __syncthreads();
            for (int w = 128; w > 0; w >>= 1) { if (t < w) sred[t] += sred[t + w]; __syncthreads(); }
            const int total = sred[0];
            __syncthreads();
            if (total >= SPARSITY) klo = mid; else khi = mid - 1u;
        }
        const unsigned tstar = klo;

        // count strictly-greater, remaining slots go to ties
        int cg = 0;
        for (int i = t; i < N_DIM; i += 256) {
            unsigned key = __float_as_uint(row[i]) & 0x7fffffffu;
            cg += (key > tstar);
        }
        sred[t] = cg; __syncthreads();
        for (int w = 128; w > 0; w >>= 1) { if (t < w) sred[t] += sred[t + w]; __syncthreads(); }
        const int c1 = sred[0];
        if (t == 0) sctr = 0u;
        __syncthreads();
        const unsigned need = (unsigned)(SPARSITY - c1);

        float* xrow = x + (size_t)b * N_DIM;
        for (int i = t; i < N_DIM; i += 256) {
            const float v = row[i];
            const unsigned key = __float_as_uint(v) & 0x7fffffffu;
            float outv = 0.0f;
            if (key > tstar) outv = v;
            else if (key == tstar) {
                unsigned slot = atomicAdd(&sctr, 1u);
                if (slot < need) outv = v;
            }
            xrow[i] = outv;
        }
    }
    __syncthreads();
    if (t == 0 && norm2 < TOL * TOL) done[b] = 1;
}

// ---------------------------------------------------------------------------
extern "C" void kernel_launch(void* const* d_in, const int* in_sizes, int n_in,
                              void* d_out, int out_size, void* d_ws, size_t ws_size,
                              hipStream_t stream) {
    const float* y = (const float*)d_in[0];   // measurements [32, 2048]
    const float* A = (const float*)d_in[1];   // A [2048, 16384]
    float* x = (float*)d_out;                 // output x [32, 16384]

    // workspace layout (floats): Rpart | R | xh | done
    float* Rpart = (float*)d_ws;                                  // KSPLIT*32*2048
    float* R     = Rpart + (size_t)KSPLIT * BATCH * M_DIM;        // 32*2048
    float* xh    = R + (size_t)BATCH * M_DIM;                     // 32*16384
    int*   done  = (int*)(xh + (size_t)BATCH * N_DIM);            // 32

    iht_init<<<(BATCH * N_DIM + 255) / 256, 256, 0, stream>>>(x, done);

    for (int it = 0; it < MAX_ITERS; it++) {
        iht_xat  <<<128 * KSPLIT, 32, 0, stream>>>(x, A, Rpart);
        iht_resid<<<(BATCH * M_DIM + 255) / 256, 256, 0, stream>>>(y, Rpart, R);
        iht_grad <<<N_DIM / 16, 32, 0, stream>>>(R, A, x, xh);
        iht_topk <<<BATCH, 256, 0, stream>>>(R, xh, x, done);
    }
}